// DGCNNFeatureExtractor_29970281791931
// MI455X (gfx1250) — compile-verified
//
#include <hip/hip_runtime.h>

typedef __attribute__((ext_vector_type(2))) float v2f;
typedef __attribute__((ext_vector_type(8))) float v8f;

#define BATCH 4
#define NPTS  4096
#define KNN   20
#define LN_EPS    1e-5f
#define NEG_SLOPE 0.2f
#define NEG_INF  -3.4e38f

__device__ __forceinline__ float wave_sum_f32(float v) {
#pragma unroll
  for (int m = 16; m > 0; m >>= 1) v += __shfl_xor(v, m, 32);
  return v;
}

// ---------------------------------------------------------------------------
// Kernel 1: center points, scale by max radius; emit padded (B,N,4) transposed
// coords and per-point squared norms.
// ---------------------------------------------------------------------------
__global__ __launch_bounds__(256)
void normalize_kernel(const float* __restrict__ x, float* __restrict__ xt,
                      float* __restrict__ xx) {
  const int b = blockIdx.x;
  const int t = threadIdx.x;
  const int NT = 256;
  __shared__ float red[256];
  __shared__ float meanv[3];
  __shared__ float sscale;
  const float* xb = x + (size_t)b * 3 * NPTS;

  float s0 = 0.f, s1 = 0.f, s2 = 0.f;
  for (int n = t; n < NPTS; n += NT) {
    s0 += xb[n]; s1 += xb[NPTS + n]; s2 += xb[2 * NPTS + n];
  }
  float sv[3] = {s0, s1, s2};
  for (int c = 0; c < 3; ++c) {
    red[t] = sv[c];
    __syncthreads();
    for (int o = 128; o > 0; o >>= 1) {
      if (t < o) red[t] += red[t + o];
      __syncthreads();
    }
    if (t == 0) meanv[c] = red[0] / (float)NPTS;
    __syncthreads();
  }
  float mx = 0.f;
  for (int n = t; n < NPTS; n += NT) {
    float d0 = xb[n] - meanv[0];
    float d1 = xb[NPTS + n] - meanv[1];
    float d2 = xb[2 * NPTS + n] - meanv[2];
    mx = fmaxf(mx, d0 * d0 + d1 * d1 + d2 * d2);
  }
  red[t] = mx;
  __syncthreads();
  for (int o = 128; o > 0; o >>= 1) {
    if (t < o) red[t] = fmaxf(red[t], red[t + o]);
    __syncthreads();
  }
  if (t == 0) sscale = 1.f / (sqrtf(red[0]) + 1e-8f);
  __syncthreads();
  const float sc = sscale;
  for (int n = t; n < NPTS; n += NT) {
    float d0 = (xb[n] - meanv[0]) * sc;
    float d1 = (xb[NPTS + n] - meanv[1]) * sc;
    float d2 = (xb[2 * NPTS + n] - meanv[2]) * sc;
    size_t o = ((size_t)b * NPTS + n) * 4;
    xt[o + 0] = d0; xt[o + 1] = d1; xt[o + 2] = d2; xt[o + 3] = 0.f;
    xx[(size_t)b * NPTS + n] = d0 * d0 + d1 * d1 + d2 * d2;
  }
}

// ---------------------------------------------------------------------------
// Kernel 2: kNN (top-20 largest pd = -||qi-cj||^2) via f32 WMMA inner products.
// One wave per 16 query rows. Each iteration computes a 16x32 pd strip with two
// independent WMMA chains (pipelines the matrix unit), then ALL 32 lanes do
// top-k insertion: lanes 0-15 scan the even 16-col sub-tile, lanes 16-31 the
// odd one, each keeping a private sorted top-20 in registers. The two per-row
// lists are merged once at the end (lower-index tie-break, matching top_k).
// ---------------------------------------------------------------------------
template <int CP>
__global__ __launch_bounds__(32)
void knn_kernel(const float* __restrict__ xq, const float* __restrict__ xx,
                int* __restrict__ idx) {
  constexpr int NS = CP / 4;
  const int lane = threadIdx.x;
  const int tilesPerB = NPTS / 16;
  const int b    = blockIdx.x / tilesPerB;
  const int row0 = (blockIdx.x % tilesPerB) * 16;
  const float* xb  = xq + (size_t)b * NPTS * CP;
  const float* xxb = xx + (size_t)b * NPTS;

  __shared__ float pd[16 * 32];
  __shared__ float mv[2][16][KNN];
  __shared__ int   mi[2][16][KNN];

  const int m    = lane & 15;        // row (A) / col (B) within tile
  const int kh   = (lane >> 4) << 1; // K sub-pair selector {0,1} vs {2,3}
  const int hh   = (lane >> 4) * 8;  // D layout: VGPR r -> row hh+r
  const int half = lane >> 4;        // sub-tile scanned in selection phase

  // Query tile in WMMA-A (16x4 f32) layout, one chain step per 4 features.
  v2f a[NS];
#pragma unroll
  for (int s = 0; s < NS; ++s) {
    const float* p = xb + (size_t)(row0 + m) * CP + 4 * s + kh;
    a[s].x = p[0];
    a[s].y = p[1];
  }
  // Query squared norms for the 8 rows this lane produces (register-resident).
  float xxq_r[8];
#pragma unroll
  for (int r = 0; r < 8; ++r) xxq_r[r] = xxb[row0 + hh + r];

  float tv[KNN];
  int   ti[KNN];
#pragma unroll
  for (int t = 0; t < KNN; ++t) { tv[t] = NEG_INF; ti[t] = 0; }

  for (int j0 = 0; j0 < NPTS; j0 += 32) {
    // Two candidate tiles in WMMA-B (4x16 f32) layout; independent chains.
    v8f acc0 = {0.f, 0.f, 0.f, 0.f, 0.f, 0.f, 0.f, 0.f};
    v8f acc1 = {0.f, 0.f, 0.f, 0.f, 0.f, 0.f, 0.f, 0.f};
    const float* cp0 = xb + (size_t)(j0 + m) * CP + kh;
    const float* cp1 = cp0 + (size_t)16 * CP;
#pragma unroll
    for (int s = 0; s < NS; ++s) {
      v2f b0; b0.x = cp0[4 * s]; b0.y = cp0[4 * s + 1];
      v2f b1; b1.x = cp1[4 * s]; b1.y = cp1[4 * s + 1];
      acc0 = __builtin_amdgcn_wmma_f32_16x16x4_f32(false, a[s], false, b0,
                                                   (short)0, acc0, false, false);
      acc1 = __builtin_amdgcn_wmma_f32_16x16x4_f32(false, a[s], false, b1,
                                                   (short)0, acc1, false, false);
    }
    const float xxj0 = xxb[j0 + m];
    const float xxj1 = xxb[j0 + 16 + m];
#pragma unroll
    for (int r = 0; r < 8; ++r) {
      pd[(hh + r) * 32 + m]      = 2.f * acc0[r] - xxq_r[r] - xxj0;
      pd[(hh + r) * 32 + 16 + m] = 2.f * acc1[r] - xxq_r[r] - xxj1;
    }
    __syncthreads();
    // Full-EXEC selection: lane l and lane l+16 both own row (l&15).
    const int cb = half * 16;
    for (int c = 0; c < 16; ++c) {
      float v = pd[m * 32 + cb + c];
      if (v > tv[KNN - 1]) {          // guarded sorted insertion (registers)
        tv[KNN - 1] = v;
        ti[KNN - 1] = j0 + cb + c;
#pragma unroll
        for (int t = KNN - 1; t > 0; --t) {
          if (tv[t] > tv[t - 1]) {
            float fv = tv[t]; tv[t] = tv[t - 1]; tv[t - 1] = fv;
            int   iv = ti[t]; ti[t] = ti[t - 1]; ti[t - 1] = iv;
          }
        }
      }
    }
    __syncthreads();
  }

  // Dump both per-row lists and merge (descending, lower index on ties).
#pragma unroll
  for (int t = 0; t < KNN; ++t) { mv[half][m][t] = tv[t]; mi[half][m][t] = ti[t]; }
  __syncthreads();
  if (lane < 16) {
    int ia = 0, ib = 0;
    int* op = idx + ((size_t)b * NPTS + row0 + lane) * KNN;
    for (int t = 0; t < KNN; ++t) {
      float va = mv[0][lane][ia];
      float vb = mv[1][lane][ib];
      int   ja = mi[0][lane][ia];
      int   jb = mi[1][lane][ib];
      bool ta = (va > vb) || ((va == vb) && (ja < jb));
      op[t] = ta ? ja : jb;
      if (ta) ++ia; else ++ib;
    }
  }
}

// ---------------------------------------------------------------------------
// Kernel 3: edge conv. One wave per point; lanes split output channels;
// LayerNorm over COUT via wave reductions; max over k neighbors.
// Emits transposed (B,N,COUT) features + per-point squared norms.
// ---------------------------------------------------------------------------
template <int C, int CP, int COUT>
__global__ __launch_bounds__(128)
void edgeconv_kernel(const float* __restrict__ xin, const int* __restrict__ idx,
                     const float* __restrict__ W, const float* __restrict__ g,
                     const float* __restrict__ be, float* __restrict__ xout,
                     float* __restrict__ xxout) {
  constexpr int CH = (COUT + 31) / 32;              // channels per lane
  constexpr int LE = (COUT < 32) ? COUT : 32;       // distinct lanes used
  __shared__ float Wl[COUT * 2 * C];
  __shared__ float gl[COUT];
  __shared__ float bl[COUT];
  for (int i = threadIdx.x; i < COUT * 2 * C; i += 128) Wl[i] = W[i];
  for (int i = threadIdx.x; i < COUT; i += 128) { gl[i] = g[i]; bl[i] = be[i]; }
  __syncthreads();

  const int lane = threadIdx.x & 31;
  const int wv   = threadIdx.x >> 5;
  const int p = blockIdx.x * 4 + wv;                // point id, grid = B*N/4
  const int b = p / NPTS;
  const int n = p % NPTS;
  const float* xb = xin + (size_t)b * NPTS * CP;

  int oc[CH];
#pragma unroll
  for (int u = 0; u < CH; ++u) oc[u] = (lane % LE) + u * 32;

  const float ctrv = (lane < C) ? xb[(size_t)n * CP + lane] : 0.f;

  // hc_o = sum_c W[o][C+c] * ctr[c]   (center half of comb, k-invariant)
  float hc[CH];
#pragma unroll
  for (int u = 0; u < CH; ++u) {
    float a = 0.f;
    for (int c = 0; c < C; ++c)
      a += Wl[oc[u] * 2 * C + C + c] * __shfl(ctrv, c, 32);
    hc[u] = a;
  }

  float mh[CH];
#pragma unroll
  for (int u = 0; u < CH; ++u) mh[u] = NEG_INF;

  const int* id = idx + ((size_t)b * NPTS + n) * KNN;
  for (int k = 0; k < KNN; ++k) {
    const int j = id[k];
    const float dv = (lane < C) ? (xb[(size_t)j * CP + lane] - ctrv) : 0.f;
    float h[CH];
    float part = 0.f;
#pragma unroll
    for (int u = 0; u < CH; ++u) {
      float a = hc[u];
      for (int c = 0; c < C; ++c)
        a += Wl[oc[u] * 2 * C + c] * __shfl(dv, c, 32);
      h[u] = a;
      part += a;
    }
    const float mean = wave_sum_f32(part) * (1.f / (32.f * CH));
    float vp = 0.f;
#pragma unroll
    for (int u = 0; u < CH; ++u) { float z = h[u] - mean; vp += z * z; }
    const float var = wave_sum_f32(vp) * (1.f / (32.f * CH));
    const float rs = rsqrtf(var + LN_EPS);
#pragma unroll
    for (int u = 0; u < CH; ++u) {
      float z = (h[u] - mean) * rs * gl[oc[u]] + bl[oc[u]];
      z = (z >= 0.f) ? z : NEG_SLOPE * z;
      mh[u] = fmaxf(mh[u], z);
    }
  }

  if (lane < LE) {
#pragma unroll
    for (int u = 0; u < CH; ++u)
      xout[((size_t)b * NPTS + n) * COUT + oc[u]] = mh[u];
  }
  float s2p = 0.f;
#pragma unroll
  for (int u = 0; u < CH; ++u) s2p += mh[u] * mh[u];
  const float s2 = wave_sum_f32(s2p) * ((float)COUT / (32.f * CH));
  if (lane == 0) xxout[(size_t)b * NPTS + n] = s2;
}

// ---------------------------------------------------------------------------
// Kernel 4: per-channel max over N into out[b*128 + off + c].
// ---------------------------------------------------------------------------
template <int CO>
__global__ __launch_bounds__(256)
void colmax_kernel(const float* __restrict__ xt, float* __restrict__ out,
                   int chanOff) {
  const int gid = blockIdx.x * blockDim.x + threadIdx.x;
  if (gid >= BATCH * CO) return;
  const int b = gid / CO;
  const int c = gid % CO;
  const float* p = xt + (size_t)b * NPTS * CO + c;
  float m = NEG_INF;
  for (int n = 0; n < NPTS; ++n) m = fmaxf(m, p[(size_t)n * CO]);
  out[b * 128 + chanOff + c] = m;
}

// ---------------------------------------------------------------------------
extern "C" void kernel_launch(void* const* d_in, const int* in_sizes, int n_in,
                              void* d_out, int out_size, void* d_ws,
                              size_t ws_size, hipStream_t stream) {
  (void)in_sizes; (void)n_in; (void)out_size; (void)ws_size;
  const float* x  = (const float*)d_in[0];
  const float* W1 = (const float*)d_in[1];
  const float* g1 = (const float*)d_in[2];
  const float* b1 = (const float*)d_in[3];
  const float* W2 = (const float*)d_in[4];
  const float* g2 = (const float*)d_in[5];
  const float* b2 = (const float*)d_in[6];
  const float* W3 = (const float*)d_in[7];
  const float* g3 = (const float*)d_in[8];
  const float* b3 = (const float*)d_in[9];
  const float* W4 = (const float*)d_in[10];
  const float* g4 = (const float*)d_in[11];
  const float* b4 = (const float*)d_in[12];
  float* out = (float*)d_out;

  char* ws = (char*)d_ws;
  auto carve = [&](size_t bytes) -> char* {
    char* p = ws;
    ws += (bytes + 255) & ~(size_t)255;
    return p;
  };
  float* xt0 = (float*)carve((size_t)BATCH * NPTS * 4 * sizeof(float));
  float* xx0 = (float*)carve((size_t)BATCH * NPTS * sizeof(float));
  int*   idx = (int*)carve((size_t)BATCH * NPTS * KNN * sizeof(int));
  float* x1t = (float*)carve((size_t)BATCH * NPTS * 16 * sizeof(float));
  float* xx1 = (float*)carve((size_t)BATCH * NPTS * sizeof(float));
  float* x2t = (float*)carve((size_t)BATCH * NPTS * 16 * sizeof(float));
  float* xx2 = (float*)carve((size_t)BATCH * NPTS * sizeof(float));
  float* x3t = (float*)carve((size_t)BATCH * NPTS * 32 * sizeof(float));
  float* xx3 = (float*)carve((size_t)BATCH * NPTS * sizeof(float));
  float* x4t = (float*)carve((size_t)BATCH * NPTS * 64 * sizeof(float));
  float* xx4 = (float*)carve((size_t)BATCH * NPTS * sizeof(float));

  const int knnGrid = BATCH * NPTS / 16;   // 1024 waves
  const int ecGrid  = BATCH * NPTS / 4;    // 4 waves / block

  normalize_kernel<<<BATCH, 256, 0, stream>>>(x, xt0, xx0);

  knn_kernel<4><<<knnGrid, 32, 0, stream>>>(xt0, xx0, idx);
  edgeconv_kernel<3, 4, 16><<<ecGrid, 128, 0, stream>>>(xt0, idx, W1, g1, b1, x1t, xx1);

  knn_kernel<16><<<knnGrid, 32, 0, stream>>>(x1t, xx1, idx);
  edgeconv_kernel<16, 16, 16><<<ecGrid, 128, 0, stream>>>(x1t, idx, W2, g2, b2, x2t, xx2);

  knn_kernel<16><<<knnGrid, 32, 0, stream>>>(x2t, xx2, idx);
  edgeconv_kernel<16, 16, 32><<<ecGrid, 128, 0, stream>>>(x2t, idx, W3, g3, b3, x3t, xx3);

  knn_kernel<32><<<knnGrid, 32, 0, stream>>>(x3t, xx3, idx);
  edgeconv_kernel<32, 32, 64><<<ecGrid, 128, 0, stream>>>(x3t, idx, W4, g4, b4, x4t, xx4);

  colmax_kernel<16><<<1, 256, 0, stream>>>(x1t, out, 0);
  colmax_kernel<16><<<1, 256, 0, stream>>>(x2t, out, 16);
  colmax_kernel<32><<<1, 256, 0, stream>>>(x3t, out, 32);
  colmax_kernel<64><<<1, 256, 0, stream>>>(x4t, out, 64);
}